// MaskedBCELoss_41566693491287
// MI455X (gfx1250) — compile-verified
//
#include <hip/hip_runtime.h>
#include <stdint.h>

// Problem constants (fixed by the reference: B=4096, L=10000, top-50)
#define L_DIM 10000
#define NV4   2500          // L / 4 float4 units per row
#define TILE  256           // float4s per tile (one per thread)
#define NT    10            // ceil(NV4 / TILE)
#define B_DIM 4096
#define KTOP  50
#define EPSF  1e-8f

typedef __attribute__((ext_vector_type(2))) float v2f;
typedef __attribute__((ext_vector_type(8))) float v8f;

// Monotonic float<->u32 mapping: key order == float order (handles negatives).
__device__ __forceinline__ unsigned fmap(float f) {
  unsigned u = __float_as_uint(f);
  return (u & 0x80000000u) ? ~u : (u | 0x80000000u);
}
__device__ __forceinline__ float funmap(unsigned k) {
  unsigned u = (k & 0x80000000u) ? (k & 0x7fffffffu) : ~k;
  return __uint_as_float(u);
}

// BCE with the reference's epsilon stabilization.
// p = sigmoid(x); 1-p computed as e*p (no cancellation).
__device__ __forceinline__ float bce_f(float x, float t) {
  float e = __expf(-x);
  float p = 1.0f / (1.0f + e);
  float q = e * p;                 // 1 - p
  return -t * __logf(p + EPSF) - (1.0f - t) * __logf(q + EPSF);
}

// CDNA5 async global->LDS copy: each lane moves 16B for logits and 16B for
// targets. VDST operand = LDS byte address in a VGPR; tracked by ASYNCcnt.
__device__ __forceinline__ void async_tile(uint64_t gL, uint64_t gT,
                                           unsigned lL, unsigned lT) {
  asm volatile("global_load_async_to_lds_b128 %0, %2, off\n\t"
               "global_load_async_to_lds_b128 %1, %3, off"
               :: "v"(lL), "v"(lT), "v"(gL), "v"(gT) : "memory");
}

__global__ void __launch_bounds__(256)
bce_topk_rows(const float* __restrict__ logits,
              const float* __restrict__ targets,
              float* __restrict__ rowmean) {
  __shared__ float4   s_stgL[2][TILE];     // 8 KB  double-buffered logits tiles
  __shared__ float4   s_stgT[2][TILE];     // 8 KB  double-buffered target tiles
  __shared__ uint4    s_key4[NV4];         // 40 KB mapped BCE keys for the row
  __shared__ unsigned s_whist[4][256];     // 4 KB  privatized histograms
  __shared__ unsigned s_bcast[2];          // {prefix, remaining-k}

  const int tid = threadIdx.x;
  const int row = blockIdx.x;
  const float* rowL = logits  + (size_t)row * L_DIM;
  const float* rowT = targets + (size_t)row * L_DIM;
  unsigned* s_key = reinterpret_cast<unsigned*>(s_key4);

  auto issue = [&](int tile, int buf) {
    int v = tile * TILE + tid;
    if (v > NV4 - 1) v = NV4 - 1;          // clamp tail: dup loads, own LDS slot
    uint64_t gL = (uint64_t)(uintptr_t)(rowL + 4 * v);
    uint64_t gT = (uint64_t)(uintptr_t)(rowT + 4 * v);
    unsigned lL = (unsigned)(uintptr_t)&s_stgL[buf][tid];
    unsigned lT = (unsigned)(uintptr_t)&s_stgT[buf][tid];
    async_tile(gL, gT, lL, lT);
  };

  // ---- Phase 1: stream row through LDS, compute BCE keys ----
  issue(0, 0);
  for (int t = 0; t < NT; ++t) {
    const int cur = t & 1;
    if (t + 1 < NT) {
      issue(t + 1, cur ^ 1);               // prefetch next tile
      asm volatile("s_wait_asynccnt 2" ::: "memory");  // tile t complete
    } else {
      asm volatile("s_wait_asynccnt 0" ::: "memory");
    }
    __syncthreads();
    int v = t * TILE + tid;
    if (v < NV4) {
      float4 x4 = s_stgL[cur][tid];
      float4 t4 = s_stgT[cur][tid];
      uint4 k;
      k.x = fmap(bce_f(x4.x, t4.x));
      k.y = fmap(bce_f(x4.y, t4.y));
      k.z = fmap(bce_f(x4.z, t4.z));
      k.w = fmap(bce_f(x4.w, t4.w));
      s_key4[v] = k;
    }
    __syncthreads();                       // protect buffer reuse next iter
  }

  // ---- Phase 2: exact top-50 threshold via MSB radix select ----
  unsigned prefix = 0;
  unsigned kk = KTOP;                      // remaining rank, in [1, 50]
  #pragma unroll
  for (int shift = 24; shift >= 0; shift -= 8) {
    s_whist[0][tid] = 0; s_whist[1][tid] = 0;
    s_whist[2][tid] = 0; s_whist[3][tid] = 0;
    __syncthreads();
    const unsigned pmask = (shift == 24) ? 0u : (0xffffffffu << (shift + 8));
    unsigned* hist = s_whist[tid >> 6];    // 4 private copies (2 waves each)
    for (int i = tid; i < L_DIM; i += 256) {
      unsigned key = s_key[i];
      if ((key & pmask) == prefix)
        atomicAdd(&hist[(key >> shift) & 0xffu], 1u);
    }
    __syncthreads();
    if (tid < 32) {                        // wave 0: suffix-scan bin select
      unsigned h[8], suf[8];
      #pragma unroll
      for (int i = 0; i < 8; ++i) {
        int b = tid * 8 + i;
        h[i] = s_whist[0][b] + s_whist[1][b] + s_whist[2][b] + s_whist[3][b];
      }
      suf[7] = h[7];
      #pragma unroll
      for (int i = 6; i >= 0; --i) suf[i] = h[i] + suf[i + 1];
      unsigned S = suf[0];                 // inclusive suffix-sum over chunks
      #pragma unroll
      for (int off = 1; off < 32; off <<= 1) {
        unsigned o = __shfl_down(S, off, 32);
        if (tid + off < 32) S += o;
      }
      unsigned above = S - suf[0];         // count in bins strictly above chunk
      #pragma unroll
      for (int i = 0; i < 8; ++i) {
        unsigned g  = above + suf[i];                      // count >= bin b
        unsigned gn = (i < 7) ? (above + suf[i + 1]) : above;  // count > bin b
        if (g >= kk && gn < kk) {          // unique selected bin
          s_bcast[0] = prefix | ((unsigned)(tid * 8 + i) << shift);
          s_bcast[1] = kk - gn;
        }
      }
    }
    __syncthreads();
    prefix = s_bcast[0];
    kk     = s_bcast[1];
  }

  // ---- Phase 3: sum of strict-greater + tie handling, row mean ----
  const unsigned K = prefix;               // exact 50th-largest key
  float local = 0.0f;
  for (int i = tid; i < L_DIM; i += 256) {
    unsigned key = s_key[i];
    if (key > K) local += funmap(key);
  }
  float* s_red = reinterpret_cast<float*>(s_whist);  // reuse histogram LDS
  s_red[tid] = local;
  __syncthreads();
  for (int off = 128; off > 0; off >>= 1) {
    if (tid < off) s_red[tid] += s_red[tid + off];
    __syncthreads();
  }
  if (tid == 0)
    rowmean[row] = (s_red[0] + (float)kk * funmap(K)) * (1.0f / KTOP);
}

// Final reduction of 4096 row means by a single wave, using the matrix pipe:
// D = A x ones(4x16): each lane's partial lands in A's K=0/K=2 slots, so
// D[m][n] = p_m + p_(m+16); summing the 8 accumulator VGPRs + one shfl_xor(16)
// gives the wave-wide total. Deterministic (fixed dataflow).
__global__ void __launch_bounds__(32)
reduce_rows(const float* __restrict__ rowmean, float* __restrict__ out) {
  const int lane = threadIdx.x;
  float p = 0.0f;
  for (int i = lane; i < B_DIM; i += 32) p += rowmean[i];

  v2f a; a[0] = p;    a[1] = 0.0f;
  v2f b; b[0] = 1.0f; b[1] = 1.0f;
  v8f c = {};
  c = __builtin_amdgcn_wmma_f32_16x16x4_f32(
      /*neg_a=*/false, a, /*neg_b=*/false, b,
      /*c_mod=*/(short)0, c, /*reuse_a=*/false, /*reuse_b=*/false);
  float s = c[0] + c[1] + c[2] + c[3] + c[4] + c[5] + c[6] + c[7];
  s += __shfl_xor(s, 16, 32);
  if (lane == 0) out[0] = s * (1.0f / (float)B_DIM);
}

extern "C" void kernel_launch(void* const* d_in, const int* in_sizes, int n_in,
                              void* d_out, int out_size, void* d_ws, size_t ws_size,
                              hipStream_t stream) {
  (void)in_sizes; (void)n_in; (void)out_size; (void)ws_size;
  const float* logits  = (const float*)d_in[0];
  const float* targets = (const float*)d_in[1];
  float* rowmeanbuf = (float*)d_ws;      // 4096 floats of scratch
  float* out = (float*)d_out;

  bce_topk_rows<<<dim3(B_DIM), dim3(256), 0, stream>>>(logits, targets, rowmeanbuf);
  reduce_rows<<<dim3(1), dim3(32), 0, stream>>>(rowmeanbuf, out);
}